// PolyFit_87514253624132
// MI455X (gfx1250) — compile-verified
//
#include <hip/hip_runtime.h>

typedef __attribute__((ext_vector_type(2))) float v2f;
typedef __attribute__((ext_vector_type(8))) float v8f;

#define DIM        512
#define DEGP1      8
#define BATCH      65536
#define ROWS_PER_WAVE  16
#define WAVES_PER_BLK  8
#define ROWS_PER_BLK   (ROWS_PER_WAVE * WAVES_PER_BLK)  // 128

// Degree-7 Horner with 8 contiguous coefficients (c[0] = x^0 term), plus the
// reference's x-bias term.
__device__ __forceinline__ float horner8(float x, const float* __restrict__ c) {
    float4 lo = *(const float4*)(c);       // c0..c3  (ds_load_b128)
    float4 hi = *(const float4*)(c + 4);   // c4..c7  (ds_load_b128)
    float v = fmaf(hi.w, x, hi.z);
    v = fmaf(v, x, hi.y);
    v = fmaf(v, x, hi.x);
    v = fmaf(v, x, lo.w);
    v = fmaf(v, x, lo.z);
    v = fmaf(v, x, lo.y);
    v = fmaf(v, x, lo.x);
    return v + x;                          // use_xbias
}

__global__ __launch_bounds__(256) void polyfit_kernel(
    const float* __restrict__ x,     // [B, 512]
    const float* __restrict__ kern,  // [8, 512, 1] flat: kern[i*512 + j]
    float* __restrict__ out) {       // [B, 1]
    // Stage the 16 KB coefficient table in LDS, transposed so the 8
    // coefficients of column j are contiguous: sk[j*8 + i].
    __shared__ float sk[DIM * DEGP1];
    for (int t = threadIdx.x; t < DIM * DEGP1; t += 256) {
        int i = t >> 9;          // t / 512
        int j = t & (DIM - 1);   // t % 512
        sk[j * DEGP1 + i] = kern[t];
    }
    __syncthreads();

    const int lane = threadIdx.x & 31;
    const int wave = threadIdx.x >> 5;
    const int r    = lane & 15;   // row within the wave's 16-row tile
    const int half = lane >> 4;   // which half of the 512 columns this lane owns

    const int  tileBase = blockIdx.x * ROWS_PER_BLK + wave * ROWS_PER_WAVE;
    const long row      = (long)tileBase + r;

    const float* __restrict__ xrow = x + row * (long)DIM + half * (DIM / 2);
    const float* __restrict__ skh  = sk + half * (DIM / 2) * DEGP1;

    // Two partial sums per lane -> exactly the 16x4 f32 WMMA A-matrix layout:
    //   lane m     : A[m,0]=a0, A[m,1]=a1   (columns [0,256))
    //   lane m+16  : A[m,2]=a0, A[m,3]=a1   (columns [256,512))
    float a0 = 0.0f, a1 = 0.0f;

#pragma unroll 4
    for (int c = 0; c < DIM / 2; c += 4) {
        float4 xv = *(const float4*)(xrow + c);  // coalesced global_load_b128
        a0 += horner8(xv.x, skh + (c + 0) * DEGP1);
        a0 += horner8(xv.y, skh + (c + 1) * DEGP1);
        a1 += horner8(xv.z, skh + (c + 2) * DEGP1);
        a1 += horner8(xv.w, skh + (c + 3) * DEGP1);
    }

    // Row-sum across lanes via one v_wmma_f32_16x16x4_f32 against an all-ones
    // B (4x16): D[m,n] = A[m,0]+A[m,1]+A[m,2]+A[m,3] for every n.
    v2f A;  A.x = a0; A.y = a1;
    v2f Bones = {1.0f, 1.0f};
    v8f Cz = {};
    v8f Dm = __builtin_amdgcn_wmma_f32_16x16x4_f32(
        /*neg_a=*/false, A, /*neg_b=*/false, Bones,
        /*c_mod=*/(short)0, Cz, /*reuse_a=*/false, /*reuse_b=*/false);

    // D-matrix f32 layout: VGPR r holds M=r on lanes 0-15 and M=r+8 on lanes
    // 16-31. Every column n is identical, so lane 0 owns rows 0..7 and lane 16
    // owns rows 8..15 of this tile.
    if ((lane & 15) == 0) {
        float* o = out + tileBase + half * 8;   // 32-byte aligned
        *(v8f*)o = Dm;
    }
}

extern "C" void kernel_launch(void* const* d_in, const int* in_sizes, int n_in,
                              void* d_out, int out_size, void* d_ws, size_t ws_size,
                              hipStream_t stream) {
    const float* x    = (const float*)d_in[0];   // [65536, 512] f32
    const float* kern = (const float*)d_in[1];   // [8, 512, 1] f32
    float*       out  = (float*)d_out;           // [65536, 1] f32

    dim3 block(256);                       // 8 waves of 32
    dim3 grid(BATCH / ROWS_PER_BLK);       // 512 blocks
    polyfit_kernel<<<grid, block, 0, stream>>>(x, kern, out);
}